// TopK_with_h_26938034880818
// MI455X (gfx1250) — compile-verified
//
#include <hip/hip_runtime.h>
#include <hip/hip_bf16.h>
#include <math.h>

// Problem constants (from reference)
#define Bsz 256
#define Nn  400
#define Dd  512
#define Kk  20

// Flat float offsets into d_out (tuple concatenated in return order)
#define OFF_OUT   0            // [B,K,K] = 102400
#define OFF_POL   102400       // [B]     = 256
#define OFF_SCR   102656       // [B,D]   = 131072
#define OFF_ENT   233728       // [B]     = 256
#define OFF_IDX   233984       // [B,K]   = 5120  (indices written as floats)

typedef __attribute__((ext_vector_type(2))) float v2f;
typedef __attribute__((ext_vector_type(4))) float v4f;
typedef __attribute__((ext_vector_type(8))) float v8f;

// ---------------------------------------------------------------------------
// Kernel 1: scorer = tanh(h_t @ W_map^T + b_map)   [256,512]=[256,512]@[512,512]^T
// FP32 WMMA: V_WMMA_F32_16X16X4_F32, one wave per 16x16 output tile.
// C[m,n] = sum_k h_t[m,k] * W_map[n,k]  (W_map^T supplies B[k,n] = W_map[n,k])
// ---------------------------------------------------------------------------
__global__ __launch_bounds__(32)
void scorer_wmma_kernel(const float* __restrict__ ht,
                        const float* __restrict__ W,
                        const float* __restrict__ bmap,
                        float* __restrict__ scorer) {
  const int tile = blockIdx.x;        // 512 tiles: 16 (M) x 32 (N)
  const int tm = tile >> 5;           // m-tile 0..15
  const int tn = tile & 31;           // n-tile 0..31
  const int lane = threadIdx.x;       // wave32
  const int half = lane >> 4;         // 0 or 1
  const int l16  = lane & 15;
  const int m0 = tm * 16, n0 = tn * 16;

  // A 16x4 fp32 layout: lane(0-15)->M, VGPR0 = K(2*half), VGPR1 = K(2*half+1)
  const float* aRow = ht + (size_t)(m0 + l16) * Dd;
  // B 4x16 fp32 layout (mirror of A): lane(0-15)->N, K = 2*half + {0,1}
  const float* bRow = W  + (size_t)(n0 + l16) * Dd;

  v8f c = {};
  for (int k0 = 0; k0 < Dd; k0 += 4) {
    const int k = k0 + 2 * half;
    v2f a; a.x = aRow[k]; a.y = aRow[k + 1];
    v2f b; b.x = bRow[k]; b.y = bRow[k + 1];
    c = __builtin_amdgcn_wmma_f32_16x16x4_f32(false, a, false, b,
                                              (short)0, c, false, false);
  }

  // C/D 16x16 fp32 layout: VGPR v holds M = v + 8*half, N = l16
  const float bias = bmap[n0 + l16];
  #pragma unroll
  for (int v = 0; v < 8; ++v) {
    const int m = m0 + v + 8 * half;
    const int n = n0 + l16;
    scorer[(size_t)m * Dd + n] = tanhf(c[v] + bias);
  }
}

// ---------------------------------------------------------------------------
// Kernel 2: invnorm[b] = 1/||scorer[b]||  (one wave per batch row)
// ---------------------------------------------------------------------------
__global__ __launch_bounds__(32)
void norm_kernel(const float* __restrict__ scorer, float* __restrict__ invn) {
  const int b = blockIdx.x;
  const int lane = threadIdx.x;
  float s = 0.f;
  #pragma unroll
  for (int t = 0; t < 16; ++t) {
    const float v = scorer[(size_t)b * Dd + lane + 32 * t];
    s += v * v;
  }
  #pragma unroll
  for (int o = 16; o > 0; o >>= 1) s += __shfl_xor(s, o, 32);
  if (lane == 0) invn[b] = 1.0f / sqrtf(s);
}

// ---------------------------------------------------------------------------
// Kernel 3: scores[b,n] = dot(node_embs[b,n,:], scorer[b,:]) * invnorm[b]
// HBM-bound: 210 MB streamed once. b128 non-temporal loads, scorer held in
// 4 float4 registers per lane (no LDS traffic), wave shuffle reduction.
// grid = (B, 4): each block owns 100 rows of one batch element.
// ---------------------------------------------------------------------------
__global__ __launch_bounds__(256)
void scores_kernel(const float* __restrict__ node,
                   const float* __restrict__ scorer,
                   const float* __restrict__ invn,
                   float* __restrict__ scores) {
  const int b    = blockIdx.x;
  const int q    = blockIdx.y;            // row quarter
  const int lane = threadIdx.x & 31;
  const int wave = threadIdx.x >> 5;      // 8 waves

  const v4f* sc4 = (const v4f*)(scorer + (size_t)b * Dd);
  const v4f s0 = sc4[lane];
  const v4f s1 = sc4[lane + 32];
  const v4f s2 = sc4[lane + 64];
  const v4f s3 = sc4[lane + 96];
  const float inv = invn[b];

  const v4f* base = (const v4f*)(node + (size_t)b * Nn * Dd);
  const int rEnd = 100 * (q + 1);
  for (int row = 100 * q + wave; row < rEnd; row += 8) {
    const v4f* r4 = base + (size_t)row * (Dd / 4);
    const v4f a0 = __builtin_nontemporal_load(r4 + lane);
    const v4f a1 = __builtin_nontemporal_load(r4 + lane + 32);
    const v4f a2 = __builtin_nontemporal_load(r4 + lane + 64);
    const v4f a3 = __builtin_nontemporal_load(r4 + lane + 96);
    float acc = a0.x*s0.x + a0.y*s0.y + a0.z*s0.z + a0.w*s0.w;
    acc      += a1.x*s1.x + a1.y*s1.y + a1.z*s1.z + a1.w*s1.w;
    acc      += a2.x*s2.x + a2.y*s2.y + a2.z*s2.z + a2.w*s2.w;
    acc      += a3.x*s3.x + a3.y*s3.y + a3.z*s3.z + a3.w*s3.w;
    #pragma unroll
    for (int o = 16; o > 0; o >>= 1) acc += __shfl_xor(acc, o, 32);
    if (lane == 0) scores[(size_t)b * Nn + row] = acc * inv;
  }
}

// ---------------------------------------------------------------------------
// Kernel 4: per-batch-row softmax stats, entropy, top-20 (tie-break: lowest
// index, matching jax.lax.top_k), gathered/scaled output, policy score.
// One block (256 threads) per batch element.
// ---------------------------------------------------------------------------
__global__ __launch_bounds__(256)
void finalize_kernel(const float* __restrict__ scores,
                     const float* __restrict__ node,
                     float* __restrict__ out) {
  const int b   = blockIdx.x;
  const int tid = threadIdx.x;
  const float NEG = -3.402823466e38f;

  __shared__ float s_s[Nn];
  __shared__ float s_w[Nn];
  __shared__ float s_red[256];
  __shared__ int   s_redi[256];
  __shared__ float s_val[Kk];
  __shared__ int   s_idx[Kk];

  for (int i = tid; i < Nn; i += 256) {
    const float v = scores[(size_t)b * Nn + i];
    s_s[i] = v;
    s_w[i] = v;
  }
  __syncthreads();

  // max
  float m = NEG;
  for (int i = tid; i < Nn; i += 256) m = fmaxf(m, s_s[i]);
  s_red[tid] = m; __syncthreads();
  for (int s = 128; s > 0; s >>= 1) {
    if (tid < s) s_red[tid] = fmaxf(s_red[tid], s_red[tid + s]);
    __syncthreads();
  }
  const float M = s_red[0]; __syncthreads();

  // sum exp
  float e = 0.f;
  for (int i = tid; i < Nn; i += 256) e += expf(s_s[i] - M);
  s_red[tid] = e; __syncthreads();
  for (int s = 128; s > 0; s >>= 1) {
    if (tid < s) s_red[tid] += s_red[tid + s];
    __syncthreads();
  }
  const float logZ = M + logf(s_red[0]); __syncthreads();

  // entropy = -sum p*logp
  float en = 0.f;
  for (int i = tid; i < Nn; i += 256) {
    const float lp = s_s[i] - logZ;
    en += expf(lp) * lp;
  }
  s_red[tid] = en; __syncthreads();
  for (int s = 128; s > 0; s >>= 1) {
    if (tid < s) s_red[tid] += s_red[tid + s];
    __syncthreads();
  }
  const float entropy = -s_red[0]; __syncthreads();

  // top-K via iterative argmax (value desc, index asc on ties)
  for (int r = 0; r < Kk; ++r) {
    float bv = NEG; int bi = 0x7fffffff;
    for (int i = tid; i < Nn; i += 256) {
      const float v = s_w[i];
      if (v > bv || (v == bv && i < bi)) { bv = v; bi = i; }
    }
    s_red[tid] = bv; s_redi[tid] = bi; __syncthreads();
    for (int s = 128; s > 0; s >>= 1) {
      if (tid < s) {
        const float ov = s_red[tid + s]; const int oi = s_redi[tid + s];
        if (ov > s_red[tid] || (ov == s_red[tid] && oi < s_redi[tid])) {
          s_red[tid] = ov; s_redi[tid] = oi;
        }
      }
      __syncthreads();
    }
    if (tid == 0) {
      s_val[r] = s_red[0];
      s_idx[r] = s_redi[0];
      s_w[s_redi[0]] = NEG;        // mask out for next round
    }
    __syncthreads();
  }

  // out[b, f, r] = node_embs[b, idx[r], f] * tanh(score[idx[r]])   (K*K = 400)
  if (tid < Kk * Kk) {
    const int f = tid / Kk;
    const int r = tid % Kk;
    const int idx = s_idx[r];
    const float t = tanhf(s_val[r]);
    out[OFF_OUT + (size_t)b * (Kk * Kk) + f * Kk + r] =
        node[((size_t)b * Nn + idx) * Dd + f] * t;
  }
  if (tid < Kk) {
    out[OFF_IDX + (size_t)b * Kk + tid] = (float)s_idx[tid];
  }
  if (tid == 0) {
    float ps = 0.f;
    #pragma unroll
    for (int r = 0; r < Kk; ++r) ps += s_val[r] - logZ;
    out[OFF_POL + b] = ps / (float)Kk;
    out[OFF_ENT + b] = entropy;
  }
}

// ---------------------------------------------------------------------------
extern "C" void kernel_launch(void* const* d_in, const int* in_sizes, int n_in,
                              void* d_out, int out_size, void* d_ws, size_t ws_size,
                              hipStream_t stream) {
  const float* node = (const float*)d_in[0];   // [B,N,D]
  // d_in[1] = mask (unused)
  const float* h_t  = (const float*)d_in[2];   // [B,D]
  const float* W    = (const float*)d_in[3];   // [D,D]
  const float* bmap = (const float*)d_in[4];   // [D]

  float* out    = (float*)d_out;
  float* scorer = out + OFF_SCR;               // scorer IS an output slice
  float* scores = (float*)d_ws;                // [B,N]
  float* invn   = scores + Bsz * Nn;           // [B]

  // 1) fp32 WMMA GEMM + tanh -> scorer (512 tiles of 16x16, one wave each)
  scorer_wmma_kernel<<<dim3(512), dim3(32), 0, stream>>>(h_t, W, bmap, scorer);
  // 2) per-row inverse norms
  norm_kernel<<<dim3(Bsz), dim3(32), 0, stream>>>(scorer, invn);
  // 3) bandwidth-bound batched mat-vec (210 MB stream)
  scores_kernel<<<dim3(Bsz, 4), dim3(256), 0, stream>>>(node, scorer, invn, scores);
  // 4) softmax stats / entropy / top-K / gathers / outputs
  finalize_kernel<<<dim3(Bsz), dim3(256), 0, stream>>>(scores, node, out);
}